// OccupancyGrid_2439541424396
// MI455X (gfx1250) — compile-verified
//
#include <hip/hip_runtime.h>

// OccupancyGrid update for MI455X (gfx1250, wave32).
// Phase 1: zero 16MB (sum,count)-pair accumulator in d_ws (RT stores -> L2 resident).
// Phase 2: NT streaming b128 loads + interleaved non-returning global_atomic_add_f32.
//          Atomics resolve in the 192MB L2; NT hints on the 128MB sample stream keep
//          the 16MB accumulator from being evicted by single-use lines.
// Phase 3: mean -> EMA -> threshold; RT reads of accumulator, NT stores of outputs.

constexpr int RES  = 128;
constexpr int NVOX = RES * RES * RES;          // 2,097,152 voxels
constexpr int SPT  = 8;                        // samples per thread in scatter

typedef float v4f __attribute__((ext_vector_type(4)));

__device__ __forceinline__ int voxel_coord(float p) {
    // reference: xn = clip((p - (-1.5)) / 3.0, 0, 1); c = int(xn*128); clip 0..127
    float f = (p + 1.5f) / 3.0f;
    f = fminf(fmaxf(f, 0.0f), 1.0f);
    int c = (int)(f * 128.0f);                 // f in [0,1] -> trunc == floor
    return c > (RES - 1) ? (RES - 1) : c;
}

__global__ void og_zero(v4f* __restrict__ ws) {
    int i = blockIdx.x * blockDim.x + threadIdx.x;   // NVOX/2 v4f total
    ws[i] = (v4f){0.0f, 0.0f, 0.0f, 0.0f};           // RT store: warm L2 lines
}

__global__ void og_scatter(const float* __restrict__ pos,
                           const float* __restrict__ den,
                           float* __restrict__ acc,      // interleaved [sum,count] per voxel
                           int n) {
    const int t = blockIdx.x * blockDim.x + threadIdx.x;
    const long long base = (long long)t * SPT;

    if (base + SPT <= n) {
        // 8 samples: 24 position floats = 6 x global_load_b128 NT (16B aligned:
        // base is a multiple of 8 -> byte offset multiple of 96).
        float pbuf[SPT * 3];
        const v4f* p4 = (const v4f*)(pos + base * 3);
#pragma unroll
        for (int k = 0; k < 6; ++k)
            ((v4f*)pbuf)[k] = __builtin_nontemporal_load(&p4[k]);

        float dv[SPT];
        const v4f* d4 = (const v4f*)(den + base);
        ((v4f*)dv)[0] = __builtin_nontemporal_load(&d4[0]);
        ((v4f*)dv)[1] = __builtin_nontemporal_load(&d4[1]);

        // Prefetch the stream ahead (next block's slice for this lane)
        // -> global_prefetch_b8, speculative.
        __builtin_prefetch(pos + base * 3 + (long long)SPT * 3 * 256, 0, 0);
        __builtin_prefetch(den + base + (long long)SPT * 256, 0, 0);

#pragma unroll
        for (int j = 0; j < SPT; ++j) {
            int cx = voxel_coord(pbuf[3 * j + 0]);
            int cy = voxel_coord(pbuf[3 * j + 1]);
            int cz = voxel_coord(pbuf[3 * j + 2]);
            int flat = (cx * RES + cy) * RES + cz;
            // Interleaved pair -> both RMWs land in the same L2 line.
            atomicAdd(&acc[2 * flat + 0], dv[j]);   // density sum
            atomicAdd(&acc[2 * flat + 1], 1.0f);    // sample count
        }
    } else {
        for (long long j = base; j < (long long)n; ++j) {
            int cx = voxel_coord(pos[3 * j + 0]);
            int cy = voxel_coord(pos[3 * j + 1]);
            int cz = voxel_coord(pos[3 * j + 2]);
            int flat = (cx * RES + cy) * RES + cz;
            atomicAdd(&acc[2 * flat + 0], den[j]);
            atomicAdd(&acc[2 * flat + 1], 1.0f);
        }
    }
}

__global__ void og_finalize(const v4f* __restrict__ acc,     // pairs (L2-hot, RT)
                            const v4f* __restrict__ ema_in,  // 4 voxels / v4f (stream)
                            v4f* __restrict__ out_grid,
                            v4f* __restrict__ out_ema) {
    int i = blockIdx.x * blockDim.x + threadIdx.x;  // voxels 4i..4i+3
    v4f a = acc[2 * i + 0];   // sum0,cnt0,sum1,cnt1  (RT: hits L2 from scatter)
    v4f b = acc[2 * i + 1];   // sum2,cnt2,sum3,cnt3
    v4f e = __builtin_nontemporal_load(&ema_in[i]);

    // mean = counts>0 ? sum/count : 0   (counts are exact small integers in f32)
    float m0 = (a.y > 0.0f) ? (a.x / a.y) : 0.0f;
    float m1 = (a.w > 0.0f) ? (a.z / a.w) : 0.0f;
    float m2 = (b.y > 0.0f) ? (b.x / b.y) : 0.0f;
    float m3 = (b.w > 0.0f) ? (b.z / b.w) : 0.0f;

    // EMA: tau=0.95, (1-tau)=0.05 (f32-rounded, matching JAX's f32 constants)
    v4f en = {0.95f * e.x + 0.05f * m0,
              0.95f * e.y + 0.05f * m1,
              0.95f * e.z + 0.05f * m2,
              0.95f * e.w + 0.05f * m3};
    v4f g  = {en.x > 0.01f ? 1.0f : 0.0f,
              en.y > 0.01f ? 1.0f : 0.0f,
              en.z > 0.01f ? 1.0f : 0.0f,
              en.w > 0.01f ? 1.0f : 0.0f};
    // Outputs are never re-read on device -> NT stores (no L2 dirty-retention).
    __builtin_nontemporal_store(g,  &out_grid[i]);   // grid_new  (output 0)
    __builtin_nontemporal_store(en, &out_ema[i]);    // grid_ema_new (output 1)
}

extern "C" void kernel_launch(void* const* d_in, const int* in_sizes, int n_in,
                              void* d_out, int out_size, void* d_ws, size_t ws_size,
                              hipStream_t stream) {
    const float* pos = (const float*)d_in[0];   // [N,3] f32
    const float* den = (const float*)d_in[1];   // [N]   f32
    const float* ema = (const float*)d_in[2];   // [1,128,128,128] f32
    const int n = in_sizes[1];                  // N samples

    float* acc = (float*)d_ws;                  // 2*NVOX floats = 16 MB

    // 1) zero accumulator: NVOX/2 v4f elements
    {
        const int total = NVOX / 2;             // 1,048,576
        og_zero<<<total / 256, 256, 0, stream>>>((v4f*)acc);
    }
    // 2) scatter histogram
    {
        const int threads = (n + SPT - 1) / SPT;
        const int blocks  = (threads + 255) / 256;
        og_scatter<<<blocks, 256, 0, stream>>>(pos, den, acc, n);
    }
    // 3) finalize: NVOX/4 voxel-quads
    {
        const int total = NVOX / 4;             // 524,288
        float* out_grid = (float*)d_out;
        float* out_ema  = (float*)d_out + NVOX;
        og_finalize<<<total / 256, 256, 0, stream>>>(
            (const v4f*)acc, (const v4f*)ema,
            (v4f*)out_grid, (v4f*)out_ema);
    }
}